// Aggregation0_44633300140708
// MI455X (gfx1250) — compile-verified
//
#include <hip/hip_runtime.h>
#include <hip/hip_bf16.h>

// Problem constants (fixed by reference setup_inputs / PATCH_W)
#define IMAGES  8
#define HORF    64
#define VERF    25
#define KP      5
#define HPIX    64      // pixels_h == pixels_w == 64
#define OW      60      // 64 - 5 + 1
#define PATCHES 3600    // 60*60

#ifndef __has_builtin
#define __has_builtin(x) 0
#endif

#if __has_builtin(__builtin_amdgcn_global_load_async_to_lds_b32)
#define HAVE_ASYNC_LDS 1
#else
#define HAVE_ASYNC_LDS 0
#endif

// One dword, per-lane arbitrary global addr -> per-lane arbitrary LDS addr.
// CDNA5 async path (GLOBAL_LOAD_ASYNC_TO_LDS_B32, tracked by ASYNCcnt);
// falls back to a plain load+ds_store if the builtin is absent.
static __device__ __forceinline__ void async_copy_dword(const float* g, float* l) {
#if HAVE_ASYNC_LDS
  __builtin_amdgcn_global_load_async_to_lds_b32(
      (__attribute__((address_space(1))) int*)(g),
      (__attribute__((address_space(3))) int*)(l),
      /*offset=*/0, /*cpol=*/0);
#else
  *l = *g;
#endif
}

static __device__ __forceinline__ void async_wait_all() {
#if HAVE_ASYNC_LDS
#if __has_builtin(__builtin_amdgcn_s_wait_asynccnt)
  __builtin_amdgcn_s_wait_asynccnt(0);
#else
  asm volatile("s_wait_asynccnt 0" ::: "memory");
#endif
#endif
}

static __device__ __forceinline__ int imax(int a, int b) { return a > b ? a : b; }
static __device__ __forceinline__ int imin(int a, int b) { return a < b ? a : b; }

// -----------------------------------------------------------------------------
// Kernel 1: fold (col2im scatter-add done as gather) + normalize by overlap cnt
// grid = IMAGES*HPIX*2 (f-half), block = 256
// LDS tile T2: idx = (dj*68 + (q+4))*32 + fl, q in [-4,63] with zero padding at
// q<0 / q>59  =>  compute does 5 unconditional ds_loads at immediate offsets:
//   idx(dj) = (w+4)*32 + fl + dj*2144      (lane addr = C + fl: conflict-free)
// img layout: (n, h, w, f) -> f innermost for coalesced store & k2 loads
// -----------------------------------------------------------------------------
__global__ __launch_bounds__(256) void fold_kernel(const float* __restrict__ x,
                                                   float* __restrict__ img) {
  __shared__ float T2[5 * 68 * 32];  // 10880 dwords = 43.5 KB

  const int bid = blockIdx.x;
  const int fc  = bid & 1;          // f-half
  const int h   = (bid >> 1) & 63;  // pixel row
  const int n   = bid >> 7;         // image
  const int f0  = fc * 32;
  const int tid = threadIdx.x;
  const int fl    = tid & 31;       // lane f
  const int wslot = tid >> 5;       // 0..7 -> w = wslot*8 + k

  // Zero the padding borders once (q+4 in {0..3} U {64..67}); async fills only
  // the interior each row, so the zeros persist across the di loop.
  for (int t = tid; t < 5 * 8 * 32; t += 256) {
    const int flz = t & 31;
    const int u   = t >> 5;               // 0..39
    const int djz = u >> 3;               // 0..4
    const int qq  = u & 7;                // 0..7
    const int q4  = (qq < 4) ? qq : (qq + 60);
    T2[(djz * 68 + q4) * 32 + flz] = 0.0f;
  }

  const int diLo = imax(0, h - (OW - 1));
  const int diHi = imin(KP - 1, h);

  float acc[8];
#pragma unroll
  for (int k = 0; k < 8; ++k) acc[k] = 0.f;

  for (int di = diLo; di <= diHi; ++di) {
    const int prow = h - di;
    // x flat index: ((n*3600 + p)*64 + f)*25 + v,  p = prow*60+q, v = di*5+dj
    const float* xb =
        x + (((size_t)n * PATCHES + (size_t)prow * OW) * HORF + f0) * VERF + di * KP;

    __syncthreads();  // padding visible / previous di's reads complete

    // Incremental walk over t = ((dj*60)+q)*32 + flx, step 256 (flx fixed,
    // q += 8 with wrap into next dj). (dj,q) are wave-uniform.
    int dj   = 0;
    int q    = tid >> 5;                          // 0..7
    int lidx = (q + 4) * 32 + fl;                 // (dj=0, q+4)
    const float* g = xb + (size_t)q * (HORF * VERF) + (size_t)fl * VERF;
    while (dj < 5) {
      async_copy_dword(g, &T2[lidx]);
      q    += 8;
      lidx += 8 * 32;
      g    += 8 * (HORF * VERF);
      if (q >= OW) {                               // wrap to next dj
        q    -= OW;
        dj   += 1;
        lidx += 8 * 32;                            // (68-60)*32
        g    += 1 - (size_t)OW * (HORF * VERF);    // +1 dword v, back 60 patches
      }
    }
    async_wait_all();
    __syncthreads();

#pragma unroll
    for (int k = 0; k < 8; ++k) {
      const int w    = wslot * 8 + k;
      const int base = (w + 4) * 32 + fl;
      float s = 0.f;
#pragma unroll
      for (int dj2 = 0; dj2 < 5; ++dj2) {
        s += T2[base + dj2 * 2144];  // (dj*67*32): invalid (w-dj) hits zeros
      }
      acc[k] += s;
    }
  }

  const int ch = diHi - diLo + 1;  // overlap count along h
#pragma unroll
  for (int k = 0; k < 8; ++k) {
    const int w  = wslot * 8 + k;
    const int cw = imin(KP - 1, w) - imax(0, w - (OW - 1)) + 1;
    const float inv = 1.0f / (float)(ch * cw);
    img[(((size_t)n * HPIX + h) * HPIX + w) * HORF + f0 + fl] = acc[k] * inv;
  }
}

// -----------------------------------------------------------------------------
// Kernel 2: unfold (im2col gather), float4 stores over 800-float spans per patch
// grid = IMAGES*OW*2 (f-half), block = 256
// LDS strides: w-stride 33, d-stride 2120 (= 64*33+8) =>
//   bank = (8*di + 33*(j+dj) + fl) mod 64 : all 25 (di,dj) combos distinct
// -----------------------------------------------------------------------------
__global__ __launch_bounds__(256) void unfold_kernel(const float* __restrict__ img,
                                                     float* __restrict__ out) {
  __shared__ float S[4 * 2120 + 63 * 33 + 32];  // 10591 dwords = 42.4 KB

  const int bid = blockIdx.x;
  const int fc  = bid & 1;
  const int i   = (bid >> 1) % OW;   // patch row
  const int n   = bid / (OW * 2);
  const int f0  = fc * 32;
  const int tid = threadIdx.x;
  const int flx = tid & 31;

  // Load img rows i..i+4, all w, 32 f. Walk m = d*64+w (wave-uniform), step 8;
  // global address is exactly linear in m (img is (n,h,w,f) row-major).
  {
    int m    = tid >> 5;                                   // 0..7
    int lidx = m * 33 + flx;                               // d=0 initially
    const float* g =
        img + (((size_t)n * HPIX + i) * HPIX + m) * HORF + f0 + flx;
    while (m < 5 * 64) {
      async_copy_dword(g, &S[lidx]);
      m    += 8;
      g    += 8 * HORF;
      lidx += 8 * 33;
      if ((m & 63) < 8) lidx += 2120 - 64 * 33;            // wrapped into next d
    }
  }
  async_wait_all();
  __syncthreads();

  // Lane owns 4 consecutive dwords r..r+3 of the 800-float (f-half x v) span;
  // decode (f,v,di,dj) once, then sweep the 60 patches of this row.
  if (tid < 200) {
    const int r = tid * 4;
    int off[4];
#pragma unroll
    for (int kk = 0; kk < 4; ++kk) {
      const int rr  = r + kk;
      const int fl2 = rr / 25;
      const int v   = rr - fl2 * 25;
      const int di  = v / 5;
      const int dj  = v - di * 5;
      off[kk] = di * 2120 + dj * 33 + fl2;   // + (j*33) at runtime
    }
    float* ob =
        out + (((size_t)n * PATCHES + (size_t)i * OW) * HORF + f0) * VERF + r;
    int joff = 0;
    for (int j = 0; j < OW; ++j) {
      float4 val;
      val.x = S[off[0] + joff];
      val.y = S[off[1] + joff];
      val.z = S[off[2] + joff];
      val.w = S[off[3] + joff];
      *(float4*)ob = val;          // 16B aligned: f0*25=800, r%4==0
      ob   += HORF * VERF;         // next patch p -> +1600 floats
      joff += 33;
    }
  }
}

extern "C" void kernel_launch(void* const* d_in, const int* in_sizes, int n_in,
                              void* d_out, int out_size, void* d_ws, size_t ws_size,
                              hipStream_t stream) {
  (void)in_sizes; (void)n_in; (void)out_size; (void)ws_size;
  const float* x = (const float*)d_in[0];  // (8, 3600, 64, 25) f32
  // d_in[1], d_in[2] are pixels_h / pixels_w scalars; fixed at 64 by the reference.
  float* out = (float*)d_out;              // (8, 3600, 64, 25) f32
  float* img = (float*)d_ws;               // (8, 64, 64, 64) f32 = 8 MB scratch

  fold_kernel<<<dim3(IMAGES * HPIX * 2), dim3(256), 0, stream>>>(x, img);
  unfold_kernel<<<dim3(IMAGES * OW * 2), dim3(256), 0, stream>>>(img, out);
}